// DynamicReindexingRAG_19499151523985
// MI455X (gfx1250) — compile-verified
//
#include <hip/hip_runtime.h>
#include <hip/hip_bf16.h>
#include <math.h>

// ---------------------------------------------------------------------------
// Problem constants (B, N, L, D, Q) = (4, 1024, 128, 128, 64), max_steps = 10
// ---------------------------------------------------------------------------
constexpr int BB        = 4;
constexpr int NL        = 1024 * 128;   // flattened document rows per batch
constexpr int DD        = 128;
constexpr int QQ        = 64;
constexpr int STEPS     = 10;
constexpr int TILE_ROWS = 512;          // rows per workgroup in streaming pass
constexpr int TILES     = NL / TILE_ROWS; // 256 tiles per batch
constexpr int REC       = 132;          // partial record: [m, s, pad, pad, acc[128]]

typedef float v8f __attribute__((ext_vector_type(8)));
typedef float v2f __attribute__((ext_vector_type(2)));

// ---------------------------------------------------------------------------
// Kernel 0: qv[b,:] = mean over q of query[b,q,:]
// ---------------------------------------------------------------------------
__global__ void prep_kernel(const float* __restrict__ query, float* __restrict__ qv) {
    int i = threadIdx.x;              // 512 threads: (b,d)
    int b = i >> 7, d = i & 127;
    const float* qp = query + (size_t)b * QQ * DD + d;
    float s = 0.f;
    #pragma unroll 8
    for (int q = 0; q < QQ; ++q) s += qp[(size_t)q * DD];
    qv[i] = s * (1.0f / QQ);
}

// ---------------------------------------------------------------------------
// Kernel P: per batch (4 blocks x 1 wave), fp32 WMMA 16x16x4 matvecs:
//   u = Q * (Wq @ qv + bq)         (kept in LDS)
//   v = Wk^T @ u                   -> ws
//   c = bk . u                     -> ws
// B-matrix holds the vector broadcast across all 16 columns, so every column
// of the 16x16 WMMA result equals the matvec tile; we extract column 0.
// ---------------------------------------------------------------------------
__global__ void project_kernel(const float* __restrict__ Wq, const float* __restrict__ bq,
                               const float* __restrict__ Wk, const float* __restrict__ bk,
                               const float* __restrict__ qv,
                               float* __restrict__ v_out, float* __restrict__ c_out) {
    const int b    = blockIdx.x;
    const int lane = threadIdx.x;        // 32 threads (one wave)
    const int half = lane >> 4;          // A/B k-split: lanes 0-15 hold K 0,1; 16-31 hold K 2,3
    const int mrow = lane & 15;

    __shared__ float qs[DD];
    __shared__ float us[DD];

    ((float4*)qs)[lane] = ((const float4*)(qv + b * DD))[lane];
    __syncthreads();

    // ---- u = 64 * (Wq @ qv + bq) ----
    #pragma unroll
    for (int rt = 0; rt < 8; ++rt) {
        v8f c = {};
        for (int kt = 0; kt < 32; ++kt) {
            const int kbase = kt * 4 + half * 2;
            v2f a, bv;
            a.x  = Wq[(rt * 16 + mrow) * DD + kbase];
            a.y  = Wq[(rt * 16 + mrow) * DD + kbase + 1];
            bv.x = qs[kbase];
            bv.y = qs[kbase + 1];
            c = __builtin_amdgcn_wmma_f32_16x16x4_f32(false, a, false, bv,
                                                      (short)0, c, false, false);
        }
        // D layout: VGPR j, lanes 0-15 -> (M=j, N=lane); lanes 16-31 -> (M=j+8).
        if (mrow == 0) {
            #pragma unroll
            for (int j = 0; j < 8; ++j) {
                const int d = rt * 16 + half * 8 + j;
                us[d] = 64.0f * (c[j] + bq[d]);
            }
        }
    }
    __syncthreads();

    // ---- v = Wk^T @ u ----
    #pragma unroll
    for (int rt = 0; rt < 8; ++rt) {
        v8f c = {};
        for (int kt = 0; kt < 32; ++kt) {
            const int kbase = kt * 4 + half * 2;
            v2f a, bv;
            a.x  = Wk[(kbase)     * DD + rt * 16 + mrow];   // Wk^T[row=e, col=d]
            a.y  = Wk[(kbase + 1) * DD + rt * 16 + mrow];
            bv.x = us[kbase];
            bv.y = us[kbase + 1];
            c = __builtin_amdgcn_wmma_f32_16x16x4_f32(false, a, false, bv,
                                                      (short)0, c, false, false);
        }
        if (mrow == 0) {
            #pragma unroll
            for (int j = 0; j < 8; ++j)
                v_out[b * DD + rt * 16 + half * 8 + j] = c[j];
        }
    }

    // ---- c = bk . u ----
    float p = 0.f;
    #pragma unroll
    for (int i = 0; i < 4; ++i) p += bk[4 * lane + i] * us[4 * lane + i];
    p += __shfl_xor(p, 16, 32);
    p += __shfl_xor(p,  8, 32);
    p += __shfl_xor(p,  4, 32);
    p += __shfl_xor(p,  2, 32);
    p += __shfl_xor(p,  1, 32);
    if (threadIdx.x == 0) c_out[b] = p;
}

// ---------------------------------------------------------------------------
// Kernel M: streaming fused pass (the HBM-bound 256 MB sweep).
// Per tile of 512 doc rows: s = row.v + c, online softmax, acc += e*row.
// Each wave: 32 lanes cover the 128-float row as float4 chunks, so a single
// global_load_b128 per lane feeds BOTH the dot product and the accumulation.
// ---------------------------------------------------------------------------
__global__ void pass_kernel(const float* __restrict__ docs,
                            const float* __restrict__ vvec,
                            const float* __restrict__ cb,
                            float* __restrict__ partials) {
    const int b    = blockIdx.y;
    const int tile = blockIdx.x;
    const int lane = threadIdx.x & 31;
    const int wave = threadIdx.x >> 5;   // 8 waves / block

    const float4 v4   = ((const float4*)(vvec + b * DD))[lane];
    const float  cbias = cb[b];
    const float* base = docs + ((size_t)b * NL + (size_t)tile * TILE_ROWS) * DD;

    float  m = -INFINITY, ssum = 0.f;
    float4 acc = {0.f, 0.f, 0.f, 0.f};

    for (int r = wave; r < TILE_ROWS; r += 8) {
        const float* row = base + (size_t)r * DD;
        // stage a future iteration's cacheline (gfx1250 global_prefetch_b8)
        __builtin_prefetch(row + 4 * 8 * DD, 0, 1);

        float4 x = ((const float4*)row)[lane];
        float p = x.x * v4.x + x.y * v4.y + x.z * v4.z + x.w * v4.w;
        p += __shfl_xor(p, 16, 32);
        p += __shfl_xor(p,  8, 32);
        p += __shfl_xor(p,  4, 32);
        p += __shfl_xor(p,  2, 32);
        p += __shfl_xor(p,  1, 32);
        const float s  = p + cbias;
        const float mn = fmaxf(m, s);
        const float sc = __expf(m - mn);   // m == -inf on first row -> 0
        const float e  = __expf(s - mn);
        acc.x = acc.x * sc + e * x.x;
        acc.y = acc.y * sc + e * x.y;
        acc.z = acc.z * sc + e * x.z;
        acc.w = acc.w * sc + e * x.w;
        ssum  = ssum  * sc + e;
        m = mn;
    }

    // merge the 8 waves of this block in LDS
    __shared__ float sm[8], ss[8];
    __shared__ float sa[8][DD];
    if (lane == 0) { sm[wave] = m; ss[wave] = ssum; }
    ((float4*)sa[wave])[lane] = acc;
    __syncthreads();

    if (wave == 0) {
        float M = sm[0];
        #pragma unroll
        for (int rr = 1; rr < 8; ++rr) M = fmaxf(M, sm[rr]);
        float4 at = {0.f, 0.f, 0.f, 0.f};
        float  st = 0.f;
        #pragma unroll
        for (int rr = 0; rr < 8; ++rr) {
            const float w  = __expf(sm[rr] - M);
            const float4 a = ((const float4*)sa[rr])[lane];
            at.x += w * a.x; at.y += w * a.y; at.z += w * a.z; at.w += w * a.w;
            st   += w * ss[rr];
        }
        float* rec = partials + (size_t)(b * TILES + tile) * REC;
        if (lane == 0) { rec[0] = M; rec[1] = st; }
        ((float4*)(rec + 4))[lane] = at;
    }
}

// ---------------------------------------------------------------------------
// Kernel R: per batch, deterministically fold 256 tile partials into
// out[b, step, :] and update qv[b] = 0.5*(qv[b] + out).
// ---------------------------------------------------------------------------
__global__ void reduce_kernel(const float* __restrict__ partials,
                              float* __restrict__ qv,
                              float* __restrict__ out, int step) {
    const int b = blockIdx.x;
    const int t = threadIdx.x;           // 128 threads
    const float* pb = partials + (size_t)b * TILES * REC;

    __shared__ float red[128];

    // global max over tile maxima
    float mx = -INFINITY;
    for (int r = t; r < TILES; r += 128) mx = fmaxf(mx, pb[(size_t)r * REC]);
    red[t] = mx;
    __syncthreads();
    for (int sft = 64; sft > 0; sft >>= 1) {
        if (t < sft) red[t] = fmaxf(red[t], red[t + sft]);
        __syncthreads();
    }
    const float M = red[0];
    __syncthreads();

    // global softmax denominator
    float sp = 0.f;
    for (int r = t; r < TILES; r += 128)
        sp += pb[(size_t)r * REC + 1] * __expf(pb[(size_t)r * REC] - M);
    red[t] = sp;
    __syncthreads();
    for (int sft = 64; sft > 0; sft >>= 1) {
        if (t < sft) red[t] += red[t + sft];
        __syncthreads();
    }
    const float S = red[0];

    // numerator for this thread's d-component
    float a = 0.f;
    for (int r = 0; r < TILES; ++r) {
        const float* rec = pb + (size_t)r * REC;
        a += rec[4 + t] * __expf(rec[0] - M);
    }
    const float o = a / S;
    out[(size_t)b * STEPS * DD + (size_t)step * DD + t] = o;   // (B, steps, D)
    qv[b * DD + t] = 0.5f * (qv[b * DD + t] + o);
}

// ---------------------------------------------------------------------------
extern "C" void kernel_launch(void* const* d_in, const int* in_sizes, int n_in,
                              void* d_out, int out_size, void* d_ws, size_t ws_size,
                              hipStream_t stream) {
    const float* query = (const float*)d_in[0];
    const float* docs  = (const float*)d_in[1];
    const float* Wq    = (const float*)d_in[2];
    const float* bq    = (const float*)d_in[3];
    const float* Wk    = (const float*)d_in[4];
    const float* bk    = (const float*)d_in[5];
    // d_in[6] = max_steps (== 10, fixed by the reference)
    float* out = (float*)d_out;
    float* ws  = (float*)d_ws;

    float* qv       = ws;            // 512 floats
    float* vvec     = ws + 512;      // 512 floats
    float* cb       = ws + 1024;     // 4 floats (+pad)
    float* partials = ws + 1056;     // B*TILES*REC floats (~529 KB), 16B aligned

    prep_kernel<<<1, BB * DD, 0, stream>>>(query, qv);
    for (int t = 0; t < STEPS; ++t) {
        project_kernel<<<BB, 32, 0, stream>>>(Wq, bq, Wk, bk, qv, vvec, cb);
        pass_kernel<<<dim3(TILES, BB), 256, 0, stream>>>(docs, vvec, cb, partials);
        reduce_kernel<<<BB, 128, 0, stream>>>(partials, qv, out, t);
    }
}